// TransferMatrixMethod_85186381349472
// MI455X (gfx1250) — compile-verified
//
#include <hip/hip_runtime.h>
#include <cstdint>
#include <cstddef>

#define NL      64
#define NINNER  62        // layers 1..NL-2 inclusive
#define EPS_N   1e-8f
#define EPS_Y   1e-9f

#define AS1 __attribute__((address_space(1)))
#define AS3 __attribute__((address_space(3)))

// ---- CDNA5 async global->LDS staging (ASYNCcnt path) -----------------------
// Probe round 1 confirmed the builtin exists; param 1 is AS1 (global) int*,
// param 2 is AS3 (LDS) int*. Cast accordingly (const dropped; it's a load).
__device__ __forceinline__ void async_load_f32(const float* g, float* l) {
#if defined(__gfx1250__) && __has_builtin(__builtin_amdgcn_global_load_async_to_lds_b32)
    __builtin_amdgcn_global_load_async_to_lds_b32(
        (AS1 int*)(uintptr_t)g, (AS3 int*)l, /*offset=*/0, /*cpol=*/0);
#else
    *l = *g;  // fallback: plain copy through VGPR
#endif
}

__device__ __forceinline__ void wait_async0() {
#if defined(__gfx1250__) && __has_builtin(__builtin_amdgcn_s_wait_asynccnt)
    __builtin_amdgcn_s_wait_asynccnt(0);
#elif defined(__gfx1250__)
    asm volatile("s_wait_asynccnt 0" ::: "memory");
#endif
}

// ---- kernel ----------------------------------------------------------------
// One block = one batch row b, 256 wavelengths. 8 wave32 waves per block.
__global__ __launch_bounds__(256)
void tmm_kernel(const float* __restrict__ n_layers,
                const float* __restrict__ d_layers,
                const float* __restrict__ wavelengths,
                float* __restrict__ out, int Wtot)
{
    __shared__ float  s_n[NL];
    __shared__ float  s_d[NL];
    __shared__ float4 s_layer[NINNER];   // {n*d, n, 1/(n+eps), 0}

    const int t = threadIdx.x;
    const int b = blockIdx.y;
    const float* nrow = n_layers + (size_t)b * NL;
    const float* drow = d_layers + (size_t)b * NL;

    // Stage the two 64-float rows into LDS asynchronously (waves 0..3 issue).
    if (t < NL) {
        async_load_f32(nrow + t, &s_n[t]);
    } else if (t < 2 * NL) {
        async_load_f32(drow + (t - NL), &s_d[t - NL]);
    }
    wait_async0();
    __syncthreads();

    // Precompute packed per-layer params once per block.
    if (t >= 1 && t <= NINNER) {
        float nv = s_n[t];
        float dv = s_d[t];
        float4 L;
        L.x = nv * dv;                // phase factor: phi = 2*pi * (n*d) / lambda
        L.y = nv;
        L.z = 1.0f / (nv + EPS_N);
        L.w = 0.0f;
        s_layer[t - 1] = L;
    }
    __syncthreads();

    const int w = blockIdx.x * blockDim.x + t;
    if (w >= Wtot) return;

    const float n_in  = s_n[0];
    const float n_sub = s_n[NL - 1];
    // v_sin/v_cos compute sin/cos(2*pi*x): feed x = (n*d)/lambda directly —
    // no 2*pi scaling, no range reduction (x <= ~1.4 revolutions here).
    const float invl = 1.0f / wavelengths[w];

    // Running product M = [[A, -iB], [-iC, D]] (real A,B,C,D), M0 = I.
    float A = 1.0f, Bv = 0.0f, C = 0.0f, D = 1.0f;

#pragma unroll 2
    for (int i = 0; i < NINNER; ++i) {
        float4 L = s_layer[i];        // broadcast ds_load_b128 (uniform addr)
        float ph = L.x * invl;
        float s  = __builtin_amdgcn_sinf(ph);   // v_sin_f32: sin(2*pi*ph)
        float c  = __builtin_amdgcn_cosf(ph);   // v_cos_f32: cos(2*pi*ph)
        float ns = L.y * s;           // n * sin
        float sn = L.z * s;           // sin / n
        float An = fmaf(-ns, Bv, A * c);   // A*c - B*(n*s)
        float Bn = fmaf( sn, A,  Bv * c);  // A*(s/n) + B*c
        float Cn = fmaf( ns, D,  C * c);   // C*c + D*(n*s)
        float Dn = fmaf(-sn, C,  D * c);   // D*c - C*(s/n)
        A = An; Bv = Bn; C = Cn; D = Dn;
    }

    // E = A - i*B*n_sub ; H = D*n_sub - i*C ; Y = H/(E+eps)
    // r = (n_in - Y)/(n_in + Y) = (n_in*(E+eps) - H)/(n_in*(E+eps) + H)
    float ae = n_in * (A + EPS_Y);
    float bn = n_in * (Bv * n_sub);
    float dn = D * n_sub;
    float num_re = ae - dn;
    float num_im = C - bn;
    float den_re = ae + dn;
    float den_im = -bn - C;
    float R = fmaf(num_re, num_re, num_im * num_im) /
              fmaf(den_re, den_re, den_im * den_im);

    out[(size_t)b * Wtot + w] = R;
}

extern "C" void kernel_launch(void* const* d_in, const int* in_sizes, int n_in,
                              void* d_out, int out_size, void* d_ws, size_t ws_size,
                              hipStream_t stream)
{
    const float* n_layers = (const float*)d_in[0];
    const float* d_layers = (const float*)d_in[1];
    const float* wavelens = (const float*)d_in[2];
    float* out = (float*)d_out;

    const int Wtot = in_sizes[2];           // 2048
    const int Bb   = in_sizes[0] / NL;      // 1024

    dim3 block(256, 1, 1);
    dim3 grid((Wtot + 255) / 256, Bb, 1);
    hipLaunchKernelGGL(tmm_kernel, grid, block, 0, stream,
                       n_layers, d_layers, wavelens, out, Wtot);
}